// MTlayer_83700322664519
// MI455X (gfx1250) — compile-verified
//
#include <hip/hip_runtime.h>
#include <math.h>

// Problem constants (match the reference).
#define LATENCY 60
#define UNITS   512
#define BATCH   64
#define TIME    2048

#define T_CHUNK 128                       // timesteps per block
#define DEG_PER_RAD 57.29577951308232f    // 180/pi
#define LOG2E       1.4426950408889634f

// Each block: 512 threads (16 wave32s), one unit per thread, one (b, t-chunk).
// exp(-dθ²·2/θσ²)·exp(-dls²·2/sσ²) is fused into a single exp2 by folding
// log2(e) into the per-unit negative inverse-variance constants.
__global__ __launch_bounds__(UNITS) void mt_layer_kernel(
    const float* __restrict__ inputs,     // [B, T, 3]
    const float* __restrict__ amp,        // [U]
    const float* __restrict__ baseline,   // [U]
    const float* __restrict__ thetapref,  // [U]
    const float* __restrict__ thetasig,   // [U]
    const float* __restrict__ speedpref,  // [U]
    const float* __restrict__ speedsig,   // [U]
    float* __restrict__ out)              // [B, T, U]
{
    // Packed per-timestep scalars: {theta_deg, log2(speed), coh, pad}
    __shared__ float4 s_scal[T_CHUNK];

    const int tid = threadIdx.x;
    const int chunks_per_batch = TIME / T_CHUNK;
    const int b  = blockIdx.x / chunks_per_batch;
    const int t0 = (blockIdx.x % chunks_per_batch) * T_CHUNK;

    // ---- per-unit constants, held in registers (one unit per thread) ----
    const int   u   = tid;
    const float tp  = thetapref[u];
    const float ts  = thetasig[u];
    const float ss  = speedsig[u];
    const float nD  = (-2.0f * LOG2E) / (ts * ts);   // folds exp->exp2 conversion
    const float nS  = (-2.0f * LOG2E) / (ss * ss);
    const float lpp = log2f(speedpref[u]);
    const float a   = amp[u];
    const float bl  = baseline[u];

    // ---- threads 0..T_CHUNK-1 derive the shared per-timestep scalars ----
    if (tid < T_CHUNK) {
        const int t = t0 + tid;
        float vx = 0.0f, vy = 0.0f, c = 0.0f;
        if (t >= LATENCY) {
            const float* p = inputs + ((size_t)b * TIME + t) * 3;
            vx = p[0];
            vy = p[1];
            c  = p[2] * 0.01f;
        }
        const float sp2   = vx * vx + vy * vy;
        const float theta = atan2f(vx, vy) * DEG_PER_RAD;  // jnp.arctan2(vx, vy)
        // speed==0 (latency prefix): force the speed-Gaussian to underflow to
        // exactly 0 (matches exp(-inf)=0) without inf/NaN intermediates.
        const float lsp = (sp2 > 0.0f) ? log2f(sqrtf(sp2)) : -1.0e15f;
        s_scal[tid] = make_float4(theta, lsp, c, 0.0f);
    }

    // Warm L2 for the next chunk's input slice (gfx1250 global_prefetch_b8).
    if (tid == 0 && (t0 + T_CHUNK) < TIME) {
        __builtin_prefetch(inputs + ((size_t)b * TIME + t0 + T_CHUNK) * 3, 0, 1);
    }

    __syncthreads();

    // ---- stream 2 KB (512 contiguous floats) per iteration, NT stores ----
    float* op = out + ((size_t)b * TIME + t0) * UNITS + u;
    #pragma unroll 4
    for (int i = 0; i < T_CHUNK; ++i) {
        const float4 s   = s_scal[i];          // one ds_load_b128, broadcast
        const float  dth = s.x - tp;
        const float  dls = s.y - lpp;
        const float  e2  = fmaf(dls * dls, nS, dth * dth * nD);
        const float  r   = fmaf(a * s.z, exp2f(e2), bl);
        __builtin_nontemporal_store(r, op);    // 256MB stream > 192MB L2: NT
        op += UNITS;
    }
}

extern "C" void kernel_launch(void* const* d_in, const int* in_sizes, int n_in,
                              void* d_out, int out_size, void* d_ws, size_t ws_size,
                              hipStream_t stream) {
    (void)in_sizes; (void)n_in; (void)out_size; (void)d_ws; (void)ws_size;

    const float* inputs    = (const float*)d_in[0];
    const float* amp       = (const float*)d_in[1];
    const float* baseline  = (const float*)d_in[2];
    const float* thetapref = (const float*)d_in[3];
    const float* thetasig  = (const float*)d_in[4];
    const float* speedpref = (const float*)d_in[5];
    const float* speedsig  = (const float*)d_in[6];
    float* out = (float*)d_out;

    dim3 grid(BATCH * (TIME / T_CHUNK));   // 64 * 16 = 1024 blocks
    dim3 block(UNITS);                     // 512 threads = 16 wave32s

    hipLaunchKernelGGL(mt_layer_kernel, grid, block, 0, stream,
                       inputs, amp, baseline, thetapref, thetasig,
                       speedpref, speedsig, out);
}